// GCN_5634997093116
// MI455X (gfx1250) — compile-verified
//
#include <hip/hip_runtime.h>

typedef __attribute__((ext_vector_type(2))) float v2f;
typedef __attribute__((ext_vector_type(8))) float v8f;

#define N_NODES 100000
#define F_IN    256
#define F_OUT   32
#define M_TILES (N_NODES / 16)   // 6250, exact

// -------------------------------------------------------------------------
// Kernel 1: h = x @ W via V_WMMA_F32_16X16X4_F32 (full fp32 matrix pipe).
// One wave -> 16x32 output tile (two 16x16 f32 accumulators).
// W staged in LDS in K-paired layout: Wp[p][c] = {W[2p][c], W[2p+1][c]}
// so each B fragment is a single ds_load_b64 (no VGPR shuffling).
// -------------------------------------------------------------------------
__global__ __launch_bounds__(256) void gcn_gemm_xw(const float* __restrict__ x,
                                                   const float* __restrict__ W,
                                                   float* __restrict__ h) {
    __shared__ v2f Wp[(F_IN / 2) * F_OUT];   // 128 x 32 x 8B = 32 KB
    for (int idx = threadIdx.x; idx < (F_IN / 2) * F_OUT; idx += 256) {
        int p = idx >> 5;        // K-pair index
        int c = idx & 31;        // output column
        v2f w;
        w.x = W[(2 * p)     * F_OUT + c];
        w.y = W[(2 * p + 1) * F_OUT + c];
        Wp[idx] = w;
    }
    __syncthreads();

    const int wave = threadIdx.x >> 5;
    const int lane = threadIdx.x & 31;
    const int tile = blockIdx.x * 8 + wave;
    if (tile >= M_TILES) return;          // wave-uniform: EXEC stays all-1s

    const int m0   = tile << 4;
    const int mrow = m0 + (lane & 15);
    const int koff = (lane >> 4) << 1;    // lanes 0-15: K=k,k+1 ; lanes 16-31: K=k+2,k+3
    const int col  = lane & 15;
    const int poff = koff >> 1;           // K-pair offset: 0 or 1

    v8f c0 = {};   // output cols 0..15
    v8f c1 = {};   // output cols 16..31

    const float* __restrict__ xrow = x + (size_t)mrow * F_IN;
    #pragma unroll 4
    for (int k = 0; k < F_IN; k += 4) {
        // A fragment: contiguous 8B load per lane (global_load_b64)
        v2f a = *(const v2f*)(xrow + k + koff);
        // B fragments: one ds_load_b64 each, directly in operand layout
        const int p = (k >> 1) + poff;
        v2f b0 = Wp[p * F_OUT + col];
        v2f b1 = Wp[p * F_OUT + 16 + col];
        c0 = __builtin_amdgcn_wmma_f32_16x16x4_f32(false, a, false, b0,
                                                   (short)0, c0, false, false);
        c1 = __builtin_amdgcn_wmma_f32_16x16x4_f32(false, a, false, b1,
                                                   (short)0, c1, false, false);
    }

    // C/D layout: VGPR j -> row m0 + j (lanes 0-15) / m0 + 8 + j (lanes 16-31)
    const int rbase = m0 + ((lane >> 4) << 3);
    #pragma unroll
    for (int j = 0; j < 8; ++j) {
        h[(size_t)(rbase + j) * F_OUT + col]      = c0[j];
        h[(size_t)(rbase + j) * F_OUT + 16 + col] = c1[j];
    }
}

// -------------------------------------------------------------------------
// Degree / normalization pipeline
// -------------------------------------------------------------------------
__global__ void gcn_deg_init(unsigned* __restrict__ deg) {
    int i = blockIdx.x * blockDim.x + threadIdx.x;
    if (i < N_NODES) deg[i] = 1u;         // self-loop
}

__global__ void gcn_deg_count(const int* __restrict__ dst,
                              unsigned* __restrict__ deg, int E) {
    int e = blockIdx.x * blockDim.x + threadIdx.x;
    if (e < E) atomicAdd(&deg[dst[e]], 1u);
}

__global__ void gcn_dinv(const unsigned* __restrict__ deg,
                         float* __restrict__ dinv) {
    int i = blockIdx.x * blockDim.x + threadIdx.x;
    if (i < N_NODES) dinv[i] = rsqrtf((float)deg[i]);  // deg >= 1 always
}

// -------------------------------------------------------------------------
// out init: self-loop contribution + bias: out[i][f] = h[i][f]*dinv[i]^2 + b[f]
// -------------------------------------------------------------------------
__global__ void gcn_out_init(const float* __restrict__ h,
                             const float* __restrict__ dinv,
                             const float* __restrict__ b,
                             float* __restrict__ out) {
    int t = blockIdx.x * blockDim.x + threadIdx.x;
    if (t >= N_NODES * F_OUT) return;
    int i = t >> 5;
    int f = t & 31;
    float di = dinv[i];
    out[t] = h[t] * di * di + b[f];
}

// -------------------------------------------------------------------------
// Edge scatter: 32 lanes per edge (lane == feature). Coalesced 128B gather
// of h[src], coalesced fp32 atomic burst into out[dst] (resolves in L2:
// out is 12.8 MB << 192 MB L2).
// -------------------------------------------------------------------------
__global__ __launch_bounds__(256) void gcn_scatter(const int* __restrict__ src,
                                                   const int* __restrict__ dst,
                                                   const float* __restrict__ h,
                                                   const float* __restrict__ dinv,
                                                   float* __restrict__ out, int E) {
    long long t = (long long)blockIdx.x * blockDim.x + threadIdx.x;
    int e = (int)(t >> 5);
    int f = (int)(t & 31);
    if (e >= E) return;
    int s = src[e];
    int d = dst[e];
    float nrm = dinv[s] * dinv[d];
    unsafeAtomicAdd(&out[(size_t)d * F_OUT + f],
                    h[(size_t)s * F_OUT + f] * nrm);
}

// -------------------------------------------------------------------------
extern "C" void kernel_launch(void* const* d_in, const int* in_sizes, int n_in,
                              void* d_out, int out_size, void* d_ws, size_t ws_size,
                              hipStream_t stream) {
    const float* x     = (const float*)d_in[0];
    const int*   edges = (const int*)d_in[1];   // (2, E) row-major, int32
    const float* W     = (const float*)d_in[2];
    const float* b     = (const float*)d_in[3];
    float*       out   = (float*)d_out;

    const int E = in_sizes[1] / 2;
    const int* src = edges;
    const int* dst = edges + E;

    // Workspace layout
    char* ws = (char*)d_ws;
    float*    h    = (float*)ws;                               // N*32 f32 = 12.8 MB
    unsigned* deg  = (unsigned*)(ws + (size_t)N_NODES * F_OUT * 4);
    float*    dinv = (float*)(ws + (size_t)N_NODES * F_OUT * 4 + (size_t)N_NODES * 4);

    // 1) GEMM h = x @ W  (WMMA f32)
    gcn_gemm_xw<<<(M_TILES + 7) / 8, 256, 0, stream>>>(x, W, h);

    // 2) Degrees + normalization
    gcn_deg_init<<<(N_NODES + 255) / 256, 256, 0, stream>>>(deg);
    gcn_deg_count<<<(E + 255) / 256, 256, 0, stream>>>(dst, deg, E);
    gcn_dinv<<<(N_NODES + 255) / 256, 256, 0, stream>>>(deg, dinv);

    // 3) Self-loop + bias init of out
    gcn_out_init<<<(N_NODES * F_OUT + 255) / 256, 256, 0, stream>>>(h, dinv, b, out);

    // 4) Edge aggregation
    long long work = (long long)E * F_OUT;
    int blocks = (int)((work + 255) / 256);
    gcn_scatter<<<blocks, 256, 0, stream>>>(src, dst, h, dinv, out, E);
}